// Encoder_36936718745769
// MI455X (gfx1250) — compile-verified
//
#include <hip/hip_runtime.h>

typedef __attribute__((ext_vector_type(2))) float v2f;
typedef __attribute__((ext_vector_type(8))) float v8f;

#define HIDC 64
#define LDSW 68   // padded LDS row stride: (68r+c)%64=(4r+c)%64 -> conflict-free frag reads

// ---------------- degree / normalization ----------------

__global__ void init_deg_kernel(float* __restrict__ deg, int n) {
  int i = blockIdx.x * blockDim.x + threadIdx.x;
  if (i < n) deg[i] = 1.0f;  // self-loop contributes 1
}

__global__ void deg_scatter_kernel(const int* __restrict__ dst, float* __restrict__ deg, int E) {
  int e = blockIdx.x * blockDim.x + threadIdx.x;
  if (e < E) unsafeAtomicAdd(&deg[dst[e]], 1.0f);
}

__global__ void dinv_kernel(float* __restrict__ deg, int n) {
  int i = blockIdx.x * blockDim.x + threadIdx.x;
  if (i < n) {
    float d = deg[i];
    deg[i] = (d > 0.0f) ? rsqrtf(d) : 0.0f;
  }
}

// ---------------- fp32 WMMA GEMM:  H[n x 64] = A[n x K] @ W[K x 64] ----------------
// block = 256 threads = 8 waves; each wave owns one 16x16 output tile
// (2 M-subtiles x 4 N-tiles per block => 32 rows per block)

__global__ __launch_bounds__(256) void gemm_wmma_kernel(
    const float* __restrict__ A, const float* __restrict__ W,
    float* __restrict__ H, int n, int K) {
  __shared__ float tile[32 * LDSW];

  const int t     = threadIdx.x;
  const int wid   = t >> 5;
  const int lane  = t & 31;
  const int row16 = lane & 15;
  const int half  = lane >> 4;      // 0: K pair {0,1}, 1: K pair {2,3}
  const int m_sub = wid >> 2;       // 0..1
  const int n0    = (wid & 3) << 4; // 0,16,32,48
  const int m_base = blockIdx.x * 32;

  v8f acc = {};

  // cooperative-load mapping: 8 threads per row, 8 consecutive floats per thread
  const int lr = t >> 3;
  const int lc = (t & 7) << 3;

  for (int kb = 0; kb < K; kb += 64) {
    // stage A[m_base..m_base+31][kb..kb+63] into LDS (coalesced float4 pairs)
    float4 v0 = make_float4(0.f, 0.f, 0.f, 0.f);
    float4 v1 = v0;
    int grow = m_base + lr;
    if (grow < n) {
      const float* p = A + (size_t)grow * K + kb + lc;
      v0 = *(const float4*)(p);
      v1 = *(const float4*)(p + 4);
    }
    float* q = &tile[lr * LDSW + lc];
    *(float4*)(q)     = v0;
    *(float4*)(q + 4) = v1;
    __syncthreads();

    const float* arow = &tile[(m_sub * 16 + row16) * LDSW];
#pragma unroll
    for (int kk = 0; kk < 64; kk += 4) {
      // A fragment 16x4: lane(row16,half) -> K = kk + half*2 + {0,1}
      v2f a = *(const v2f*)(arow + kk + half * 2);
      // B fragment 4x16: W[krow][n0+row16], W[krow+1][n0+row16]
      int krow = kb + kk + half * 2;
      v2f b;
      b.x = W[(size_t)krow * HIDC + n0 + row16];
      b.y = W[(size_t)(krow + 1) * HIDC + n0 + row16];
      acc = __builtin_amdgcn_wmma_f32_16x16x4_f32(
          false, a, false, b, (short)0, acc, false, false);
    }
    __syncthreads();
  }

  // C/D layout: VGPR r, lanes 0-15 -> M=r, lanes 16-31 -> M=r+8
  int mrow = m_base + m_sub * 16 + half * 8;
#pragma unroll
  for (int r = 0; r < 8; ++r) {
    int gr = mrow + r;
    if (gr < n) H[(size_t)gr * HIDC + n0 + row16] = acc[r];
  }
}

// ---------------- propagation ----------------

// agg[i] = h[i]*dinv[i]^2 (+ bias)   — self-loop term, initializes accumulator
__global__ void selfloop_init_kernel(const float* __restrict__ h,
                                     const float* __restrict__ dinv,
                                     const float* __restrict__ bias,
                                     float* __restrict__ agg,
                                     int total, int use_bias) {
  int i = blockIdx.x * blockDim.x + threadIdx.x;
  if (i >= total) return;
  int node = i >> 6;
  float d = dinv[node];
  float v = h[i] * d * d;
  if (use_bias) v += bias[i & 63];
  agg[i] = v;
}

// one wave per edge: 32 lanes x float2 = full 64-channel row (256B coalesced)
__global__ void edge_scatter_kernel(const int* __restrict__ src,
                                    const int* __restrict__ dst,
                                    const float* __restrict__ dinv,
                                    const float* __restrict__ h,
                                    float* __restrict__ agg, int E) {
  int t = blockIdx.x * blockDim.x + threadIdx.x;
  int e = t >> 5;
  if (e >= E) return;
  int c = (t & 31) << 1;
  int s = src[e];
  int d = dst[e];
  float norm = dinv[s] * dinv[d];
  float2 hv = *(const float2*)(h + (size_t)s * HIDC + c);
  float* out = agg + (size_t)d * HIDC + c;
  unsafeAtomicAdd(out,     hv.x * norm);
  unsafeAtomicAdd(out + 1, hv.y * norm);
}

__global__ void bias_relu_kernel(float* __restrict__ agg,
                                 const float* __restrict__ bias, int total) {
  int i = blockIdx.x * blockDim.x + threadIdx.x;
  if (i >= total) return;
  float v = agg[i] + bias[i & 63];
  agg[i] = v > 0.0f ? v : 0.0f;
}

// ---------------- launcher ----------------

extern "C" void kernel_launch(void* const* d_in, const int* in_sizes, int n_in,
                              void* d_out, int out_size, void* d_ws, size_t ws_size,
                              hipStream_t stream) {
  const float* x  = (const float*)d_in[0];
  const int*   ei = (const int*)d_in[1];
  const float* W1 = (const float*)d_in[2];
  const float* b1 = (const float*)d_in[3];
  const float* W2 = (const float*)d_in[4];
  const float* b2 = (const float*)d_in[5];

  const int IN_CH = 512;
  const int n = in_sizes[0] / IN_CH;
  const int E = in_sizes[1] / 2;
  const int* src = ei;        // edge_index[0]
  const int* dst = ei + E;    // edge_index[1]
  float* out = (float*)d_out;

  float* ws = (float*)d_ws;
  size_t n_al = ((size_t)n + 255) & ~(size_t)255;
  float* dinv = ws;                         // [n]
  float* bufA = ws + n_al;                  // [n*64] : GEMM outputs
  float* bufB = bufA + (size_t)n * HIDC;    // [n*64] : aggregation / h1

  const int B = 256;
  const long long total = (long long)n * HIDC;
  unsigned gN   = (unsigned)((n + B - 1) / B);
  unsigned gE   = (unsigned)((E + B - 1) / B);
  unsigned gNC  = (unsigned)((total + B - 1) / B);
  unsigned gE32 = (unsigned)(((long long)E * 32 + B - 1) / B);
  unsigned gMM  = (unsigned)((n + 31) / 32);

  // normalization: deg (with self-loop) -> dinv = deg^{-1/2}
  init_deg_kernel<<<gN, B, 0, stream>>>(dinv, n);
  deg_scatter_kernel<<<gE, B, 0, stream>>>(dst, dinv, E);
  dinv_kernel<<<gN, B, 0, stream>>>(dinv, n);

  // ---- layer 1: h = x@W1 ; agg = segsum(norm * h[src]) ; h1 = relu(agg + b1)
  gemm_wmma_kernel<<<gMM, 256, 0, stream>>>(x, W1, bufA, n, IN_CH);
  selfloop_init_kernel<<<gNC, B, 0, stream>>>(bufA, dinv, b1, bufB, (int)total, 0);
  edge_scatter_kernel<<<gE32, B, 0, stream>>>(src, dst, dinv, bufA, bufB, E);
  bias_relu_kernel<<<gNC, B, 0, stream>>>(bufB, b1, (int)total);

  // ---- layer 2: h2 = h1@W2 ; out = segsum(norm * h2[src]) + b2
  gemm_wmma_kernel<<<gMM, 256, 0, stream>>>(bufB, W2, bufA, n, HIDC);
  selfloop_init_kernel<<<gNC, B, 0, stream>>>(bufA, dinv, b2, out, (int)total, 1);
  edge_scatter_kernel<<<gE32, B, 0, stream>>>(src, dst, dinv, bufA, out, E);
}